// PointeNet_60919816126535
// MI455X (gfx1250) — compile-verified
//
#include <hip/hip_runtime.h>
#include <hip/hip_bf16.h>

typedef __attribute__((ext_vector_type(2))) float v2f;
typedef __attribute__((ext_vector_type(8))) float v8f;

#define WMMA_F32X4(a, b, c) \
  __builtin_amdgcn_wmma_f32_16x16x4_f32(false, (a), false, (b), (short)0, (c), false, false)

constexpr int B_ = 4, N_ = 8192, C_ = 144, G_ = 2048, Kn = 32;
constexpr int NG = B_ * G_;          // 8192
constexpr int NTOT = NG * Kn;        // 262144 = 2^18
constexpr int NP = B_ * N_;          // 32768

// ---------------- embed: e[p][o] = w_embed[o,:3] . xyz[p,:3] ----------------
__global__ void k_embed(const float* __restrict__ xyz, const float* __restrict__ w,
                        float* __restrict__ x) {
  int p = blockIdx.x, o = threadIdx.x;
  if (o < C_) {
    const float* pt = xyz + p * 3;
    x[(size_t)p * C_ + o] = w[o * 3] * pt[0] + w[o * 3 + 1] * pt[1] + w[o * 3 + 2] * pt[2];
  }
}

// per-channel accumulation for [p][c] layout (coalesced: thread == channel)
__global__ void k_chanacc(const float* __restrict__ x, float* __restrict__ acc,
                          int np, int nc) {
  int o = threadIdx.x;
  if (o >= nc) return;
  float s = 0.f, q = 0.f;
  for (int p = blockIdx.x; p < np; p += gridDim.x) {
    float v = x[(size_t)p * nc + o];
    s += v; q += v * v;
  }
  atomicAdd(&acc[o], s);
  atomicAdd(&acc[nc + o], q);
}

__global__ void k_finstats(const float* __restrict__ acc, float* __restrict__ st,
                           int nc, float cnt) {
  int o = blockIdx.x * blockDim.x + threadIdx.x;
  if (o < nc) {
    float m = acc[o] / cnt;
    float v = acc[nc + o] / cnt - m * m;
    st[2 * o] = m;
    st[2 * o + 1] = rsqrtf(v + 1e-5f);
  }
}

__global__ void k_normrelu_flat(float* __restrict__ x, const float* __restrict__ st,
                                int total, int nc) {
  int e = blockIdx.x * 256 + threadIdx.x;
  if (e >= total) return;
  int o = e % nc;
  x[e] = fmaxf((x[e] - st[2 * o]) * st[2 * o + 1], 0.f);
}

__global__ void k_zero(float* __restrict__ p, int n) {
  int e = blockIdx.x * 256 + threadIdx.x;
  if (e < n) p[e] = 0.f;
}

// ---------------- FPS: 1 block per batch, 1024 threads, 8 pts/thread ----------------
__global__ __launch_bounds__(1024) void k_fps(const float* __restrict__ xyz,
                                              int* __restrict__ fps) {
  int b = blockIdx.x, t = threadIdx.x;
  const float* P = xyz + (size_t)b * N_ * 3;
  float px[8], py[8], pz[8], dist[8];
#pragma unroll
  for (int i = 0; i < 8; i++) {
    int n = i * 1024 + t;
    px[i] = P[n * 3]; py[i] = P[n * 3 + 1]; pz[i] = P[n * 3 + 2];
    dist[i] = 1e10f;
  }
  __shared__ float bx, by, bz;
  __shared__ float sv[32];
  __shared__ int si[32];
  if (t == 0) { fps[b * G_] = 0; bx = P[0]; by = P[1]; bz = P[2]; }
  __syncthreads();
  for (int it = 1; it < G_; ++it) {
    float cx = bx, cy = by, cz = bz;
    float bvv = -1.f; int bii = 0;
#pragma unroll
    for (int i = 0; i < 8; i++) {
      float dx = px[i] - cx, dy = py[i] - cy, dz = pz[i] - cz;
      float d = fminf(dist[i], dx * dx + dy * dy + dz * dz);
      dist[i] = d;
      int n = i * 1024 + t;
      if (d > bvv) { bvv = d; bii = n; }
    }
    for (int m = 16; m >= 1; m >>= 1) {
      float ov = __shfl_xor(bvv, m, 32);
      int oi = __shfl_xor(bii, m, 32);
      if (ov > bvv || (ov == bvv && oi < bii)) { bvv = ov; bii = oi; }
    }
    if ((t & 31) == 0) { sv[t >> 5] = bvv; si[t >> 5] = bii; }
    __syncthreads();
    if (t < 32) {
      bvv = sv[t]; bii = si[t];
      for (int m = 16; m >= 1; m >>= 1) {
        float ov = __shfl_xor(bvv, m, 32);
        int oi = __shfl_xor(bii, m, 32);
        if (ov > bvv || (ov == bvv && oi < bii)) { bvv = ov; bii = oi; }
      }
      if (t == 0) {
        fps[b * G_ + it] = bii;
        bx = P[bii * 3]; by = P[bii * 3 + 1]; bz = P[bii * 3 + 2];
      }
    }
    __syncthreads();
  }
}

// ---------------- KNN: thread per center, sorted insertion list of 32 ----------------
__global__ void k_knn(const float* __restrict__ xyz, const int* __restrict__ fps,
                      int* __restrict__ knn) {
  constexpr int TILE = 512;
  __shared__ float sp[TILE * 3];
  int gi = blockIdx.x * 256 + threadIdx.x;       // 0..8191, block-uniform batch
  int b = gi >> 11;
  const float* P = xyz + (size_t)b * N_ * 3;
  int ci = fps[gi];
  float cx = P[ci * 3], cy = P[ci * 3 + 1], cz = P[ci * 3 + 2];
  float nd[Kn]; int ni[Kn];
#pragma unroll
  for (int i = 0; i < Kn; i++) { nd[i] = 3.4e38f; ni[i] = 0; }
  for (int t0 = 0; t0 < N_; t0 += TILE) {
    __syncthreads();
    for (int i = threadIdx.x; i < TILE * 3; i += 256) sp[i] = P[t0 * 3 + i];
    __syncthreads();
    for (int j = 0; j < TILE; j++) {
      float dx = sp[j * 3] - cx, dy = sp[j * 3 + 1] - cy, dz = sp[j * 3 + 2] - cz;
      float d = dx * dx + dy * dy + dz * dz;
      if (d < nd[Kn - 1]) {
        int q = Kn - 1;
        while (q > 0 && nd[q - 1] > d) { nd[q] = nd[q - 1]; ni[q] = ni[q - 1]; --q; }
        nd[q] = d; ni[q] = t0 + j;
      }
    }
  }
  for (int i = 0; i < Kn; i++) knn[gi * Kn + i] = ni[i];
}

// ------ gather/transpose: xknn[c][n], lcx[c][bg], diff[i][n], via LDS staging ------
__global__ void k_gather(const float* __restrict__ x, const float* __restrict__ xyz,
                         const int* __restrict__ fps, const int* __restrict__ knn,
                         float* __restrict__ xknn, float* __restrict__ lcx,
                         float* __restrict__ diff) {
  __shared__ float sx[33 * C_];
  __shared__ int sidx[33];
  int bg = blockIdx.x, t = threadIdx.x, b = bg >> 11;
  if (t < Kn) sidx[t] = knn[bg * Kn + t];
  if (t == Kn) sidx[Kn] = fps[bg];
  __syncthreads();
  const float* X = x + (size_t)b * N_ * C_;
  for (int e = t; e < 33 * C_; e += 256) {
    int r = e / C_, c = e - r * C_;
    sx[e] = X[(size_t)sidx[r] * C_ + c];
  }
  __syncthreads();
  int base = bg * Kn;
  for (int e = t; e < C_ * Kn; e += 256) {
    int c = e >> 5, k = e & 31;
    xknn[(size_t)c * NTOT + base + k] = sx[k * C_ + c];
  }
  if (t < C_) lcx[(size_t)t * NG + bg] = sx[Kn * C_ + t];
  const float* P = xyz + (size_t)b * N_ * 3;
  if (t < Kn * 3) {
    int k = t / 3, i = t - k * 3;
    diff[(size_t)i * NTOT + base + k] = P[sidx[k] * 3 + i] - P[sidx[Kn] * 3 + i];
  }
}

// ---------------- normal + curvature via analytic symmetric 3x3 eigen ----------------
__global__ void k_normalcurv(const float* __restrict__ xyz, const int* __restrict__ knn,
                             float* __restrict__ nrm, float* __restrict__ cur) {
  int bg = blockIdx.x * 256 + threadIdx.x;
  if (bg >= NG) return;
  int b = bg >> 11;
  const float* P = xyz + (size_t)b * N_ * 3;
  float mx = 0, my = 0, mz = 0, xx = 0, xy = 0, xz = 0, yy = 0, yz = 0, zz = 0;
#pragma unroll
  for (int k = 0; k < 8; k++) {
    int i = knn[bg * Kn + k];
    float px = P[i * 3], py = P[i * 3 + 1], pz = P[i * 3 + 2];
    mx += px; my += py; mz += pz;
    xx += px * px; xy += px * py; xz += px * pz;
    yy += py * py; yz += py * pz; zz += pz * pz;
  }
  const float i8 = 0.125f;
  mx *= i8; my *= i8; mz *= i8;
  float a11 = xx * i8 - mx * mx, a12 = xy * i8 - mx * my, a13 = xz * i8 - mx * mz;
  float a22 = yy * i8 - my * my, a23 = yz * i8 - my * mz, a33 = zz * i8 - mz * mz;
  float q = (a11 + a22 + a33) * (1.f / 3.f);
  float p1 = a12 * a12 + a13 * a13 + a23 * a23;
  float b11 = a11 - q, b22 = a22 - q, b33 = a33 - q;
  float p2 = b11 * b11 + b22 * b22 + b33 * b33 + 2.f * p1;
  float e1 = q, e2 = q, e3 = q, nx = 1.f, ny = 0.f, nz = 0.f;
  if (p2 > 1e-24f) {
    float p = sqrtf(p2 * (1.f / 6.f)), ip = 1.f / p;
    float c11 = b11 * ip, c12 = a12 * ip, c13 = a13 * ip;
    float c22 = b22 * ip, c23 = a23 * ip, c33 = b33 * ip;
    float detB = c11 * (c22 * c33 - c23 * c23) - c12 * (c12 * c33 - c23 * c13) +
                 c13 * (c12 * c23 - c22 * c13);
    float r = fminf(1.f, fmaxf(-1.f, 0.5f * detB));
    float phi = acosf(r) * (1.f / 3.f);
    e1 = q + 2.f * p * cosf(phi);
    e3 = q + 2.f * p * cosf(phi + 2.0943951023931953f);
    e2 = 3.f * q - e1 - e3;
    float r1x = a11 - e3, r1y = a12, r1z = a13;
    float r2x = a12, r2y = a22 - e3, r2z = a23;
    float r3x = a13, r3y = a23, r3z = a33 - e3;
    float v1x = r1y * r2z - r1z * r2y, v1y = r1z * r2x - r1x * r2z, v1z = r1x * r2y - r1y * r2x;
    float v2x = r1y * r3z - r1z * r3y, v2y = r1z * r3x - r1x * r3z, v2z = r1x * r3y - r1y * r3x;
    float v3x = r2y * r3z - r2z * r3y, v3y = r2z * r3x - r2x * r3z, v3z = r2x * r3y - r2y * r3x;
    float n1 = v1x * v1x + v1y * v1y + v1z * v1z;
    float n2 = v2x * v2x + v2y * v2y + v2z * v2z;
    float n3 = v3x * v3x + v3y * v3y + v3z * v3z;
    float bn = n1; nx = v1x; ny = v1y; nz = v1z;
    if (n2 > bn) { bn = n2; nx = v2x; ny = v2y; nz = v2z; }
    if (n3 > bn) { bn = n3; nx = v3x; ny = v3y; nz = v3z; }
    if (bn < 1e-30f) { nx = 1.f; ny = 0.f; nz = 0.f; }
  }
  float inl = rsqrtf(nx * nx + ny * ny + nz * nz);
  nrm[bg * 3] = nx * inl; nrm[bg * 3 + 1] = ny * inl; nrm[bg * 3 + 2] = nz * inl;
  float s1 = e1 + 1e-9f, s2 = e2 + 1e-9f, s3 = e3 + 1e-9f;
  float si = 1.f / (s1 + s2 + s3);
  cur[bg * 3] = s1 * si; cur[bg * 3 + 1] = s2 * si; cur[bg * 3 + 2] = s3 * si;
}

// ---------------- tiny 3-wide matvec: out[o][bg] = w[o,:3] . v[bg,:3] ----------------
__global__ void k_emb3(const float* __restrict__ w, const float* __restrict__ v,
                       float* __restrict__ out) {
  int e = blockIdx.x * 256 + threadIdx.x;
  if (e >= C_ * NG) return;
  int o = e >> 13, bg = e & (NG - 1);
  out[e] = w[o * 3] * v[bg * 3] + w[o * 3 + 1] * v[bg * 3 + 1] + w[o * 3 + 2] * v[bg * 3 + 2];
}

// ---------------- channel stats for [c][n] layout: one block per channel ----------------
__global__ void k_stats_cn(const float* __restrict__ src, float* __restrict__ st, int nn) {
  __shared__ float ss[256], sq[256];
  int c = blockIdx.x, t = threadIdx.x;
  const float* p = src + (size_t)c * nn;
  float s = 0.f, q = 0.f;
  for (int i = t; i < nn; i += 256) { float v = p[i]; s += v; q += v * v; }
  ss[t] = s; sq[t] = q;
  __syncthreads();
  for (int m = 128; m > 0; m >>= 1) {
    if (t < m) { ss[t] += ss[t + m]; sq[t] += sq[t + m]; }
    __syncthreads();
  }
  if (t == 0) {
    float mean = ss[0] / nn;
    float var = sq[0] / nn - mean * mean;
    st[2 * c] = mean;
    st[2 * c + 1] = rsqrtf(var + 1e-5f);
  }
}

// ---------------- diff std (ddof=1) ----------------
__global__ void k_diffacc(const float* __restrict__ d, float* __restrict__ acc, int n) {
  __shared__ float ss[256], sq[256];
  float s = 0.f, q = 0.f;
  for (int i = blockIdx.x * 256 + threadIdx.x; i < n; i += gridDim.x * 256) {
    float v = d[i]; s += v; q += v * v;
  }
  ss[threadIdx.x] = s; sq[threadIdx.x] = q;
  __syncthreads();
  for (int m = 128; m > 0; m >>= 1) {
    if (threadIdx.x < m) { ss[threadIdx.x] += ss[threadIdx.x + m]; sq[threadIdx.x] += sq[threadIdx.x + m]; }
    __syncthreads();
  }
  if (threadIdx.x == 0) { atomicAdd(&acc[0], ss[0]); atomicAdd(&acc[1], sq[0]); }
}

__global__ void k_difffin(const float* __restrict__ acc, float* __restrict__ dstat) {
  const float n = 3.0f * NTOT;
  float mean = acc[0] / n;
  float var = (acc[1] - acc[0] * mean) / (n - 1.0f);
  dstat[0] = 1.0f / (sqrtf(fmaxf(var, 0.f)) + 1e-5f);
}

// ---------------- generic f32 WMMA GEMM: C[M][Nn] = A[M][lda](cols acol0..) * B[K][Nn] ----------------
// Wave owns a 16(M) x 64(N) tile: 4 accumulators, A fragment reused 4x per K-step.
// Strength-reduced B pointers; branch-free masked A rows; all 4 B fragments loaded
// into distinct registers BEFORE the WMMA burst so loads cluster and waits stagger;
// near-cache software prefetch of the B stream two K-steps ahead.
__global__ void k_wmma_gemm(const float* __restrict__ A, int lda, int acol0,
                            const float* __restrict__ Bm, float* __restrict__ Cm,
                            int M, int Kd, int Nn, int mtiles,
                            const float* __restrict__ addp, int addn, int addshift) {
  int wid = blockIdx.x * 8 + (threadIdx.x >> 5);
  int total = mtiles * (Nn >> 6);
  if (wid >= total) return;
  int lane = threadIdx.x & 31;
  int mt = wid % mtiles;
  int nb = wid / mtiles;
  int half = lane >> 4, lm = lane & 15;
  int row = mt * 16 + lm;
  int n0 = nb * 64 + lm;
  v8f z = {0.f, 0.f, 0.f, 0.f, 0.f, 0.f, 0.f, 0.f};
  v8f acc0 = z, acc1 = z, acc2 = z, acc3 = z;

  const float amask = (row < M) ? 1.f : 0.f;
  const int rowc = (row < M) ? row : (M - 1);
  const float* arow = A + (size_t)rowc * lda + acol0 + half * 2;
  const float* bp = Bm + (size_t)(half * 2) * Nn + n0;   // row k0+half*2
  const float* bq = bp + Nn;                              // row k0+half*2+1
  const size_t stepB = (size_t)4 * Nn;

#pragma unroll 2
  for (int k0 = 0; k0 < Kd; k0 += 4) {
    v2f a;
    a.x = amask * arow[0];
    a.y = amask * arow[1];
    arow += 4;
    __builtin_prefetch(bp + 2 * stepB, 0, 3);  // B stream, 2 K-steps ahead, near cache
    // load all four B fragments into distinct registers, then burst the WMMAs
    v2f b0, b1, b2, b3;
    b0.x = bp[0];      b0.y = bq[0];
    b1.x = bp[16];     b1.y = bq[16];
    b2.x = bp[32];     b2.y = bq[32];
    b3.x = bp[48];     b3.y = bq[48];
    acc0 = WMMA_F32X4(a, b0, acc0);
    acc1 = WMMA_F32X4(a, b1, acc1);
    acc2 = WMMA_F32X4(a, b2, acc2);
    acc3 = WMMA_F32X4(a, b3, acc3);
    bp += stepB;
    bq += stepB;
  }

  v8f acc[4] = {acc0, acc1, acc2, acc3};
#pragma unroll
  for (int j = 0; j < 4; j++) {
    int nc = nb * 64 + j * 16 + lm;
#pragma unroll
    for (int r = 0; r < 8; r++) {
      int rr = mt * 16 + half * 8 + r;
      if (rr < M) {
        float v = acc[j][r];
        if (addp) v += addp[(size_t)rr * addn + (nc >> addshift)];
        Cm[(size_t)rr * Nn + nc] = v;
      }
    }
  }
}

// ---------------- kw = (relu(bn(h)) + pe) * pe, in place on H ----------------
__global__ void k_kw(float* __restrict__ H, const float* __restrict__ st,
                     const float* __restrict__ diff, const float* __restrict__ dstat) {
  int e = blockIdx.x * 256 + threadIdx.x;   // < C_*NTOT = 37,748,736
  int c = e >> 18, n = e & (NTOT - 1);
  int i3 = c / 48;
  int jj = c - i3 * 48;
  int j = (jj < 24) ? jj : jj - 24;
  float dn = diff[(size_t)i3 * NTOT + n] * dstat[0];
  // 100 * dn / 1000^(j/24) ; ln(1000)/24 = 0.28782313662425287
  float dv = 100.f * dn * __expf(-0.28782313662425287f * (float)j);
  float s = (jj < 24) ? __sinf(dv) : __cosf(dv);
  float hn = fmaxf((H[e] - st[2 * c]) * st[2 * c + 1], 0.f);
  H[e] = (hn + s) * s;
}

// ---------------- normalize+relu for [c][2^shift] layout ----------------
__global__ void k_normrelu_cn(float* __restrict__ X, const float* __restrict__ st,
                              int shift, int total) {
  int e = blockIdx.x * 256 + threadIdx.x;
  if (e >= total) return;
  int c = e >> shift;
  X[e] = fmaxf((X[e] - st[2 * c]) * st[2 * c + 1], 0.f);
}

// ------ k-reduction + MAA combine: one wave per (c,bg) column of 32 ------
__global__ void k_reduce_f(const float* __restrict__ kw, const float* __restrict__ mb,
                           const float* __restrict__ stMB,
                           const float* __restrict__ neraw, const float* __restrict__ stNE,
                           const float* __restrict__ ceraw, const float* __restrict__ stCE,
                           const float* __restrict__ alpha, const float* __restrict__ beta,
                           float* __restrict__ f) {
  int wid = blockIdx.x * 8 + (threadIdx.x >> 5);
  int lane = threadIdx.x & 31;
  int c = wid >> 13, bg = wid & (NG - 1);
  if (c >= C_) return;
  size_t n = (size_t)c * NTOT + bg * 32 + lane;
  float v = fmaxf((mb[n] - stMB[2 * c]) * stMB[2 * c + 1] + kw[n], 0.f);
  float mx = v, sm = v;
  for (int m = 16; m >= 1; m >>= 1) {
    mx = fmaxf(mx, __shfl_xor(mx, m, 32));
    sm += __shfl_xor(sm, m, 32);
  }
  if (lane == 0) {
    float lc = mx + sm * (1.f / 32.f);
    int e = c * NG + bg;
    float ne = fmaxf((neraw[e] - stNE[2 * c]) * stNE[2 * c + 1], 0.f);
    float ce = fmaxf((ceraw[e] - stCE[2 * c]) * stCE[2 * c + 1], 0.f);
    f[e] = alpha[c] * lc + beta[c] + alpha[C_ + c] * ne + beta[C_ + c] +
           alpha[2 * C_ + c] * ce + beta[2 * C_ + c];
  }
}

// ---------------- final: d_out[b][o][g] = relu(bn(out_raw[o][bg])) ----------------
__global__ void k_out(const float* __restrict__ raw, const float* __restrict__ st,
                      float* __restrict__ out) {
  int e = blockIdx.x * 256 + threadIdx.x;
  if (e >= C_ * NG) return;
  int o = e >> 13, bg = e & (NG - 1);
  int b = bg >> 11, g = bg & (G_ - 1);
  out[(size_t)b * C_ * G_ + o * G_ + g] = fmaxf((raw[e] - st[2 * o]) * st[2 * o + 1], 0.f);
}

extern "C" void kernel_launch(void* const* d_in, const int* in_sizes, int n_in,
                              void* d_out, int out_size, void* d_ws, size_t ws_size,
                              hipStream_t stream) {
  (void)in_sizes; (void)n_in; (void)out_size; (void)ws_size;
  const float* xyz      = (const float*)d_in[0];
  const float* w_embed  = (const float*)d_in[1];
  const float* w_norm   = (const float*)d_in[2];
  const float* w_curv   = (const float*)d_in[3];
  const float* w_lin1   = (const float*)d_in[4];
  const float* w2a      = (const float*)d_in[5];
  const float* w2b      = (const float*)d_in[7];
  const float* maa_a    = (const float*)d_in[9];
  const float* maa_b    = (const float*)d_in[10];
  const float* w_maa    = (const float*)d_in[11];
  float* out = (float*)d_out;

  char* ws = (char*)d_ws;
  size_t off = 0;
  auto alloc = [&](size_t bytes) -> char* {
    char* p = ws + off;
    off += (bytes + 255) & ~(size_t)255;
    return p;
  };
  float* x      = (float*)alloc((size_t)NP * C_ * 4);
  int*   fps    = (int*)  alloc((size_t)NG * 4);
  int*   knn    = (int*)  alloc((size_t)NTOT * 4);
  float* xknn   = (float*)alloc((size_t)C_ * NTOT * 4);   // later reused as mb
  float* lcx    = (float*)alloc((size_t)C_ * NG * 4);
  float* diff   = (float*)alloc((size_t)3 * NTOT * 4);
  float* h2     = (float*)alloc((size_t)C_ * NG * 4);
  float* kw     = (float*)alloc((size_t)C_ * NTOT * 4);
  float* ma     = (float*)alloc((size_t)72 * NTOT * 4);
  float* mb     = xknn;                                    // alias: xknn dead after GEMM_H
  float* nrm    = (float*)alloc((size_t)NG * 3 * 4);
  float* cur    = (float*)alloc((size_t)NG * 3 * 4);
  float* neraw  = (float*)alloc((size_t)C_ * NG * 4);
  float* ceraw  = (float*)alloc((size_t)C_ * NG * 4);
  float* fbuf   = (float*)alloc((size_t)C_ * NG * 4);
  float* outraw = (float*)alloc((size_t)C_ * NG * 4);
  float* stE    = (float*)alloc(2 * C_ * 4);
  float* stH    = (float*)alloc(2 * C_ * 4);
  float* stMA   = (float*)alloc(2 * 72 * 4);
  float* stMB   = (float*)alloc(2 * C_ * 4);
  float* stNE   = (float*)alloc(2 * C_ * 4);
  float* stCE   = (float*)alloc(2 * C_ * 4);
  float* stOUT  = (float*)alloc(2 * C_ * 4);
  float* eacc   = (float*)alloc(2 * C_ * 4);
  float* dacc   = (float*)alloc(2 * 4);
  float* dstat  = (float*)alloc(4);

  // reset accumulators every call (graph-replay safe)
  k_zero<<<2, 256, 0, stream>>>(eacc, 2 * C_);
  k_zero<<<1, 256, 0, stream>>>(dacc, 2);

  // embed + BN(relu)
  k_embed<<<NP, 160, 0, stream>>>(xyz, w_embed, x);
  k_chanacc<<<256, 160, 0, stream>>>(x, eacc, NP, C_);
  k_finstats<<<1, 160, 0, stream>>>(eacc, stE, C_, (float)NP);
  k_normrelu_flat<<<(NP * C_) / 256, 256, 0, stream>>>(x, stE, NP * C_, C_);

  // FPS + KNN + gathers
  k_fps<<<B_, 1024, 0, stream>>>(xyz, fps);
  k_knn<<<NG / 256, 256, 0, stream>>>(xyz, fps, knn);
  k_gather<<<NG, 256, 0, stream>>>(x, xyz, fps, knn, xknn, lcx, diff);

  // normal / curvature embeddings
  k_normalcurv<<<NG / 256, 256, 0, stream>>>(xyz, knn, nrm, cur);
  k_emb3<<<(C_ * NG) / 256, 256, 0, stream>>>(w_norm, nrm, neraw);
  k_emb3<<<(C_ * NG) / 256, 256, 0, stream>>>(w_curv, cur, ceraw);
  k_stats_cn<<<C_, 256, 0, stream>>>(neraw, stNE, NG);
  k_stats_cn<<<C_, 256, 0, stream>>>(ceraw, stCE, NG);

  // diff global std (ddof=1)
  k_diffacc<<<512, 256, 0, stream>>>(diff, dacc, 3 * NTOT);
  k_difffin<<<1, 1, 0, stream>>>(dacc, dstat);

  // H = W1[:, :144] * xknn + (W1[:, 144:] * lcx) broadcast over k  (WMMA f32)
  k_wmma_gemm<<<(9 * (NG / 64)) / 8, 256, 0, stream>>>(w_lin1, 288, 144, lcx, h2,
                                                       144, 144, NG, 9, nullptr, 0, 0);
  k_wmma_gemm<<<(9 * (NTOT / 64)) / 8, 256, 0, stream>>>(w_lin1, 288, 0, xknn, kw,
                                                         144, 144, NTOT, 9, h2, NG, 5);
  k_stats_cn<<<C_, 256, 0, stream>>>(kw, stH, NTOT);
  k_kw<<<(C_ * NTOT) / 256, 256, 0, stream>>>(kw, stH, diff, dstat);

  // m = relu(bn(w2a * kw)) ; bias cancels under per-channel BN
  k_wmma_gemm<<<(5 * (NTOT / 64)) / 8, 256, 0, stream>>>(w2a, 144, 0, kw, ma,
                                                         72, 144, NTOT, 5, nullptr, 0, 0);
  k_stats_cn<<<72, 256, 0, stream>>>(ma, stMA, NTOT);
  k_normrelu_cn<<<(72 * NTOT) / 256, 256, 0, stream>>>(ma, stMA, 18, 72 * NTOT);

  // m = bn(w2b * m)
  k_wmma_gemm<<<(9 * (NTOT / 64)) / 8, 256, 0, stream>>>(w2b, 72, 0, ma, mb,
                                                         144, 72, NTOT, 9, nullptr, 0, 0);
  k_stats_cn<<<C_, 256, 0, stream>>>(mb, stMB, NTOT);

  // kw2 = relu(m + kw); lc = max_k + mean_k; f = maa combine
  k_reduce_f<<<(C_ * NG) / 8, 256, 0, stream>>>(kw, mb, stMB, neraw, stNE,
                                                ceraw, stCE, maa_a, maa_b, fbuf);

  // out = relu(bn(w_maa * f))
  k_wmma_gemm<<<(9 * (NG / 64)) / 8, 256, 0, stream>>>(w_maa, 144, 0, fbuf, outraw,
                                                       144, 144, NG, 9, nullptr, 0, 0);
  k_stats_cn<<<C_, 256, 0, stream>>>(outraw, stOUT, NG);
  k_out<<<(C_ * NG) / 256, 256, 0, stream>>>(outraw, stOUT, out);
}